// KNNSimple_11647951307123
// MI455X (gfx1250) — compile-verified
//
#include <hip/hip_runtime.h>
#include <hip/hip_bf16.h>

typedef __attribute__((ext_vector_type(2))) float v2f;
typedef __attribute__((ext_vector_type(4))) float v4f;
typedef __attribute__((ext_vector_type(8))) float v8f;

#define KNN 16

// ---------------------------------------------------------------------------
// Kernel 1: zero the 1-GiB adjacency matrix with wide non-temporal stores.
// Output bandwidth is the roofline floor (~46us at 23.3 TB/s).
// ---------------------------------------------------------------------------
__global__ __launch_bounds__(256) void knn_zero_kernel(v4f* __restrict__ out,
                                                       long long n4) {
    long long i = (long long)blockIdx.x * blockDim.x + threadIdx.x;
    const long long stride = (long long)gridDim.x * blockDim.x;
    v4f z = {0.0f, 0.0f, 0.0f, 0.0f};
    for (; i < n4; i += stride) {
        __builtin_nontemporal_store(z, &out[i]);
    }
}

// Sorted-ascending top-16 insert; static indices after inlining keep both
// arrays in VGPRs. Strict '<' + ascending-j insertion order preserves
// jax.top_k's lowest-index tie-break.
__device__ __forceinline__ void topk_insert(float (&bd)[KNN], int (&bi)[KNN],
                                            float d2, int j) {
#pragma unroll
    for (int p = KNN - 1; p >= 1; --p) {
        if (d2 < bd[p - 1])    { bd[p] = bd[p - 1]; bi[p] = bi[p - 1]; }
        else if (d2 < bd[p])   { bd[p] = d2;        bi[p] = j; }
    }
    if (d2 < bd[0]) { bd[0] = d2; bi[0] = j; }
}

// ---------------------------------------------------------------------------
// Kernel 2: per-row top-16 via V_WMMA_F32_16X16X4_F32, zero-transpose scheme.
//
// Role swap: M = streaming column tile (16/iter), N = the wave's 16 fixed
// rows, K = 4. The spare K=3 slot carries the column norm:
//     A[c,:] = (-2x_c, -2y_c, -2z_c, ||c||^2)   (per tile)
//     B[:,r] = ( x_r ,  y_r ,  z_r ,    1   )   (fixed per wave)
//     C[q]   = ||r||^2                          (lane-constant, hoisted)
// so D[q] = d^2(col = t*16 + q + 8*(lane>=16), row = rowBase + (lane&15))
// lands directly in the lane that owns that row — no LDS transpose.
//
// The column-coordinate load (one global_load_b96) is software-pipelined one
// tile ahead so its latency overlaps the WMMA + min-reduction of the current
// tile instead of sitting exposed in front of them.
//
// Lane pairs (l, l^16) cover the two column halves of one row and merge
// their sorted lists once at the end through LDS (same wave -> in-order DS).
// ---------------------------------------------------------------------------
__global__ __launch_bounds__(256) void knn_topk_kernel(
    const float* __restrict__ nodes, float* __restrict__ out, int n) {
    __shared__ float sD[256 * KNN];
    __shared__ int   sI[256 * KNN];

    const int lane = threadIdx.x & 31;
    const int wave = threadIdx.x >> 5;
    const bool hi  = lane >= 16;
    const int m    = lane & 15;

    const int rowBase = (blockIdx.x * 8 + wave) * 16;
    const int row = rowBase + m;

    // Fixed per-wave B matrix (4x16: K x rows) and C (row norms).
    const float rx = nodes[3 * row + 0];
    const float ry = nodes[3 * row + 1];
    const float rz = nodes[3 * row + 2];
    const float sqr = rx * rx + ry * ry + rz * rz;

    // B layout: lane -> N=lane&15; VGPR0 = K0(lanes 0-15)/K2(lanes 16-31),
    // VGPR1 = K1/K3.
    v2f B;
    B[0] = hi ? rz : rx;
    B[1] = hi ? 1.0f : ry;

    v8f C;
#pragma unroll
    for (int q = 0; q < 8; ++q) C[q] = sqr;

    float bd[KNN];
    int   bi[KNN];
#pragma unroll
    for (int k = 0; k < KNN; ++k) { bd[k] = 3.402823466e38f; bi[k] = 0; }

    const int colOff = hi ? 8 : 0;
    const int ntiles = n >> 4;

    // --- software pipeline prologue: coords for tile 0 ---
    float cx = nodes[3 * m + 0];
    float cy = nodes[3 * m + 1];
    float cz = nodes[3 * m + 2];

#pragma unroll 1
    for (int t = 0; t < ntiles; ++t) {
        // Issue next tile's column load now (wraps to 0 on the last tile so
        // the load is always unconditional and in-bounds); its wait lands
        // after the WMMA + reduction below.
        const int tn = (t + 1 == ntiles) ? 0 : (t + 1);
        const int cn = (tn << 4) + m;
        const float nx = nodes[3 * cn + 0];
        const float ny = nodes[3 * cn + 1];
        const float nz = nodes[3 * cn + 2];

        // Process current tile from registers.
        const float sqc = cx * cx + cy * cy + cz * cz;

        // A layout: lane -> M=lane&15; VGPR0 = K0/K2, VGPR1 = K1/K3.
        v2f A;
        A[0] = hi ? (-2.0f * cz) : (-2.0f * cx);
        A[1] = hi ? sqc          : (-2.0f * cy);

        // Finished squared-distance tile in one WMMA.
        v8f D = __builtin_amdgcn_wmma_f32_16x16x4_f32(
            false, A, false, B, (short)0, C, false, false);

        // Fast reject: only scan candidates if some d^2 beats current 16th.
        const float mn = fminf(fminf(fminf(D[0], D[1]), fminf(D[2], D[3])),
                               fminf(fminf(D[4], D[5]), fminf(D[6], D[7])));
        if (mn < bd[KNN - 1]) {
            const int jb = (t << 4) + colOff;
#pragma unroll
            for (int q = 0; q < 8; ++q) {
                const float d2 = D[q];
                const int j = jb + q;
                if (j != row && d2 < bd[KNN - 1]) topk_insert(bd, bi, d2, j);
            }
        }

        // Rotate pipeline registers.
        cx = nx; cy = ny; cz = nz;
    }

    // Merge the two half-row lists of lane pair (l, l^16) via LDS.
    float* myD = &sD[threadIdx.x * KNN];
    int*   myI = &sI[threadIdx.x * KNN];
#pragma unroll
    for (int k = 0; k < KNN; ++k) { myD[k] = bd[k]; myI[k] = bi[k]; }
    asm volatile("" ::: "memory");  // same-wave partner: DS in-order, no barrier

    const int partner = (int)threadIdx.x ^ 16;
    const float* pD = &sD[partner * KNN];
    const int*   pI = &sI[partner * KNN];
#pragma unroll 1
    for (int k = 0; k < KNN; ++k) {
        const float d2 = pD[k];
        const int j = pI[k];
        if (d2 < bd[KNN - 1]) topk_insert(bd, bi, d2, j);
    }

    // Lanes 0-15 scatter the 16 neighbor flags for their row.
    if (!hi) {
        const size_t rowOff = (size_t)row * (size_t)n;
#pragma unroll
        for (int k = 0; k < KNN; ++k) {
            out[rowOff + (size_t)bi[k]] = 1.0f;
        }
    }
}

extern "C" void kernel_launch(void* const* d_in, const int* in_sizes, int n_in,
                              void* d_out, int out_size, void* d_ws, size_t ws_size,
                              hipStream_t stream) {
    const float* nodes = (const float*)d_in[0];
    float* out = (float*)d_out;
    const int n = in_sizes[0] / 3;  // 16384

    // 1) zero 1 GiB adjacency with NT float4 stores
    const long long n4 = ((long long)n * (long long)n) / 4;
    knn_zero_kernel<<<8192, 256, 0, stream>>>((v4f*)out, n4);

    // 2) WMMA distance tiles + per-row top-16 scatter
    // 256 threads = 8 waves x 16 rows -> 128 rows per block
    knn_topk_kernel<<<n / 128, 256, 0, stream>>>(nodes, out, n);
}